// lstm_bottleneck_53317724012895
// MI455X (gfx1250) — compile-verified
//
#include <hip/hip_runtime.h>
#include <math.h>

typedef __attribute__((ext_vector_type(2))) float v2f;
typedef __attribute__((ext_vector_type(8))) float v8f;

#define BB 64
#define TT 512
#define DD 256
#define UU 512
#define GG 2048   // 4*U

__device__ __forceinline__ float sigf(float x) {
  return 1.0f / (1.0f + __expf(-x));
}

// ---------------------------------------------------------------------------
// Projection: xw[b*T+t, g] = sum_d x[b,t,d] * W[d,g] + bias[g]
// One wave computes a 16(M) x 64(N) output tile with V_WMMA_F32_16X16X4_F32.
// A 16x4 f32 frag: lanes0-15 hold M=lane (K=k0,k0+1), lanes16-31 M=lane-16
// (K=k0+2,k0+3). B 4x16 frag mirrors that in K. C/D: VGPR v -> row v (+8 for
// upper half-wave), col = lane&15.
// ---------------------------------------------------------------------------
__global__ __launch_bounds__(256) void proj_kernel(const float* __restrict__ x,
                                                   const float* __restrict__ W,
                                                   const float* __restrict__ bias,
                                                   float* __restrict__ xw) {
  const int lane  = threadIdx.x & 31;
  const int wave  = threadIdx.x >> 5;
  const int gw    = blockIdx.x * 8 + wave;   // 0..65535
  const int mtile = gw >> 5;                 // 0..2047  (M = B*T = 32768)
  const int nbase = (gw & 31) * 64;          // 0..1984
  const int half  = lane >> 4;
  const int l15   = lane & 15;
  const float* xrow = x + (size_t)(mtile * 16 + l15) * DD;

  v8f c[4];
  for (int j = 0; j < 4; ++j) {
    float bv = bias[nbase + j * 16 + l15];
    for (int v = 0; v < 8; ++v) c[j][v] = bv;
  }
  for (int k0 = 0; k0 < DD; k0 += 4) {
    float4 av = *(const float4*)(xrow + k0);
    v2f a;
    a.x = half ? av.z : av.x;
    a.y = half ? av.w : av.y;
    const int krow = k0 + half * 2;
    for (int j = 0; j < 4; ++j) {
      const int n = nbase + j * 16 + l15;
      v2f bf;
      bf.x = W[(size_t)krow * GG + n];
      bf.y = W[(size_t)(krow + 1) * GG + n];
      c[j] = __builtin_amdgcn_wmma_f32_16x16x4_f32(false, a, false, bf,
                                                   (short)0, c[j], false, false);
    }
  }
  for (int j = 0; j < 4; ++j) {
    const int n = nbase + j * 16 + l15;
    for (int v = 0; v < 8; ++v) {
      const int row = mtile * 16 + v + half * 8;
      xw[(size_t)row * GG + n] = c[j][v];
    }
  }
}

// ---------------------------------------------------------------------------
// One recurrence step. 128 waves: wave -> (mtile in 0..3, utile in 0..31).
// Each wave owns one 16x16 u-tile but accumulates ALL FOUR gate columns
// (n = gate*512 + u) so the cell update is done in-register.
// FUSED=true : z starts at bias, adds x_t@W (K=256) then h@R (K=512).
// FUSED=false: z starts at precomputed xw tile, adds h@R only.
// ---------------------------------------------------------------------------
template <bool FUSED>
__global__ __launch_bounds__(256) void step_kernel(const float* __restrict__ x,
                                                   const float* __restrict__ W,
                                                   const float* __restrict__ xw,
                                                   const float* __restrict__ R,
                                                   const float* __restrict__ bias,
                                                   const float* __restrict__ h_in,
                                                   float* __restrict__ h_out,
                                                   float* __restrict__ c_st,
                                                   int t) {
  const int lane  = threadIdx.x & 31;
  const int wave  = threadIdx.x >> 5;
  const int gw    = blockIdx.x * 8 + wave;   // 0..127
  const int mtile = gw & 3;                  // batch tile (B=64)
  const int utile = gw >> 2;                 // 0..31 (U=512)
  const int half  = lane >> 4;
  const int l15   = lane & 15;
  const int ucol  = utile * 16 + l15;

  v8f c[4];
  if (FUSED) {
    for (int g = 0; g < 4; ++g) {
      float bv = bias[g * UU + ucol];
      for (int v = 0; v < 8; ++v) c[g][v] = bv;
    }
  } else {
    for (int g = 0; g < 4; ++g)
      for (int v = 0; v < 8; ++v) {
        const int b = mtile * 16 + v + half * 8;
        c[g][v] = xw[((size_t)b * TT + t) * GG + g * UU + ucol];
      }
  }

  if (FUSED) {
    // z += x[:,t,:] @ W   (K = D = 256)
    const float* xrow = x + ((size_t)(mtile * 16 + l15) * TT + t) * DD;
    for (int k0 = 0; k0 < DD; k0 += 4) {
      float4 av = *(const float4*)(xrow + k0);
      v2f a;
      a.x = half ? av.z : av.x;
      a.y = half ? av.w : av.y;
      const int krow = k0 + half * 2;
      for (int g = 0; g < 4; ++g) {
        const int n = g * UU + ucol;
        v2f bf;
        bf.x = W[(size_t)krow * GG + n];
        bf.y = W[(size_t)(krow + 1) * GG + n];
        c[g] = __builtin_amdgcn_wmma_f32_16x16x4_f32(false, a, false, bf,
                                                     (short)0, c[g], false, false);
      }
    }
  }

  // z += h @ R   (K = U = 512)
  const float* hrow = h_in + (size_t)(mtile * 16 + l15) * UU;
  for (int k0 = 0; k0 < UU; k0 += 4) {
    float4 av = *(const float4*)(hrow + k0);
    v2f a;
    a.x = half ? av.z : av.x;
    a.y = half ? av.w : av.y;
    const int krow = k0 + half * 2;
    for (int g = 0; g < 4; ++g) {
      const int n = g * UU + ucol;
      v2f bf;
      bf.x = R[(size_t)krow * GG + n];
      bf.y = R[(size_t)(krow + 1) * GG + n];
      c[g] = __builtin_amdgcn_wmma_f32_16x16x4_f32(false, a, false, bf,
                                                   (short)0, c[g], false, false);
    }
  }

  // Gates: i, f, g(relu), o ; c' = f*c + i*g ; h' = o*relu(c')
  for (int v = 0; v < 8; ++v) {
    const int b = mtile * 16 + v + half * 8;
    const int idx = b * UU + ucol;
    float zi = c[0][v], zf = c[1][v], zg = c[2][v], zo = c[3][v];
    float i_ = sigf(zi);
    float f_ = sigf(zf);
    float g_ = fmaxf(zg, 0.0f);
    float o_ = sigf(zo);
    float cn = f_ * c_st[idx] + i_ * g_;
    c_st[idx] = cn;
    h_out[idx] = o_ * fmaxf(cn, 0.0f);
  }
}

// out[b,t,u] = h[b,u] ; float4 vectorized (U % 4 == 0)
__global__ __launch_bounds__(256) void bcast_kernel(const float* __restrict__ h,
                                                    float* __restrict__ out) {
  const size_t e = ((size_t)blockIdx.x * blockDim.x + threadIdx.x) * 4;
  const int b = (int)(e / ((size_t)TT * UU));
  const int u = (int)(e % UU);
  float4 v = *(const float4*)(h + (size_t)b * UU + u);
  *(float4*)(out + e) = v;
}

extern "C" void kernel_launch(void* const* d_in, const int* in_sizes, int n_in,
                              void* d_out, int out_size, void* d_ws, size_t ws_size,
                              hipStream_t stream) {
  const float* x    = (const float*)d_in[0];  // [64,512,256]
  const float* W    = (const float*)d_in[1];  // [256,2048]
  const float* R    = (const float*)d_in[2];  // [512,2048]
  const float* bias = (const float*)d_in[3];  // [2048]
  float* out = (float*)d_out;                 // [64,512,512]

  float* hA = (float*)d_ws;                   // [64,512]
  float* hB = hA + BB * UU;
  float* cS = hB + BB * UU;
  float* xw = cS + BB * UU;                   // [32768, 2048] if it fits

  const size_t state_bytes = (size_t)3 * BB * UU * sizeof(float);
  const size_t xw_bytes    = (size_t)BB * TT * GG * sizeof(float);
  const bool precompute = (ws_size >= state_bytes + xw_bytes);

  hipMemsetAsync(hA, 0, (size_t)BB * UU * sizeof(float), stream);
  hipMemsetAsync(cS, 0, (size_t)BB * UU * sizeof(float), stream);

  if (precompute) {
    // 65536 waves of 16x64 tiles -> 8192 blocks of 8 waves
    proj_kernel<<<8192, 256, 0, stream>>>(x, W, bias, xw);
  }

  for (int t = 0; t < TT; ++t) {
    const float* hi = (t & 1) ? hB : hA;
    float*       ho = (t & 1) ? hA : hB;
    if (precompute)
      step_kernel<false><<<16, 256, 0, stream>>>(x, W, xw, R, bias, hi, ho, cS, t);
    else
      step_kernel<true><<<16, 256, 0, stream>>>(x, W, xw, R, bias, hi, ho, cS, t);
  }

  // T=512 steps (even): final h is in hA
  bcast_kernel<<<(BB * TT * UU) / (256 * 4), 256, 0, stream>>>(hA, out);
}